// TopKSparsity_48232482734126
// MI455X (gfx1250) — compile-verified
//
#include <hip/hip_runtime.h>
#include <stdint.h>

// Row-wise top-k mask for MI455X (gfx1250, wave32).
// One 256-thread block (8 wave32s) per row. Row staged in LDS via CDNA5
// async global->LDS loads (ASYNCcnt path); exact k-th-largest found by
// 4-pass MSB radix select on order-preserving u32 keys; ties broken by
// lowest index (matches jax.lax.top_k). HBM traffic = 1 read + 1 write.

#define ROW_D      16384
#define TOPK       64
#define NT         256
#define CHUNKS     (ROW_D * 4 / 16)     // 4096 x 16B chunks per row
#define CHUNKS_PT  (CHUNKS / NT)        // 16 async b128 loads per thread
#define VEC_ITERS  (ROW_D / 4 / NT)     // 16 float4 per thread per pass
#define EQCAP      1024

// Dynamic LDS layout (bytes):
//   [0,            65536)  : row data (fp32)
//   [65536,        66560)  : 256-bin histogram (u32)
//   [66560,        66576)  : scalars: prefix, remaining, eqcount, pad
//   [66576,        70672)  : eq-index list (u32 x 1024)
#define LDS_ROW_OFF   0
#define LDS_HIST_OFF  65536
#define LDS_SC_OFF    66560
#define LDS_EQ_OFF    66576
#define LDS_BYTES     70672

typedef __attribute__((ext_vector_type(4))) float v4f;

__device__ __forceinline__ uint32_t order_key(float f) {
    uint32_t u = __float_as_uint(f);
    // monotonic float->uint map: larger float => larger key
    return u ^ ((uint32_t)((int32_t)u >> 31) | 0x80000000u);
}

__global__ __launch_bounds__(NT) void TopKSparsity_48232482734126_kernel(
    const float* __restrict__ x, float* __restrict__ out) {
    extern __shared__ char smem[];
    float*    row    = (float*)(smem + LDS_ROW_OFF);
    v4f*      row4   = (v4f*)(smem + LDS_ROW_OFF);
    uint32_t* hist   = (uint32_t*)(smem + LDS_HIST_OFF);
    uint32_t* sc     = (uint32_t*)(smem + LDS_SC_OFF);   // [0]=prefix [1]=remaining [2]=eqcount
    uint32_t* eqlist = (uint32_t*)(smem + LDS_EQ_OFF);

    const int tid = threadIdx.x;
    const float* rowp = x   + (size_t)blockIdx.x * ROW_D;
    float*       outp = out + (size_t)blockIdx.x * ROW_D;
    v4f*         outp4 = (v4f*)outp;

    // Dynamic LDS begins right after static LDS: this is row[]'s LDS byte base.
    uint32_t ldsbase = __builtin_amdgcn_groupstaticsize() + LDS_ROW_OFF;

    // ---- Stage row into LDS with CDNA5 async loads (ASYNCcnt path) ----
#pragma unroll
    for (int it = 0; it < CHUNKS_PT; ++it) {
        uint32_t c    = (uint32_t)tid + (uint32_t)it * NT;
        uint32_t voff = c * 16u;            // byte offset within row (GVS mode)
        uint32_t dst  = ldsbase + voff;     // per-lane LDS byte address (VDST)
        asm volatile("global_load_async_to_lds_b128 %0, %1, %2"
                     :: "v"(dst), "v"(voff), "s"(rowp)
                     : "memory");
    }
    asm volatile("s_wait_asynccnt 0" ::: "memory");

    if (tid == 0) { sc[0] = 0u; sc[1] = TOPK; sc[2] = 0u; }
    __syncthreads();

    // ---- 4-pass MSB radix select for the k-th largest key ----
#pragma unroll 1
    for (int pass = 0; pass < 4; ++pass) {
        hist[tid] = 0u;                     // NT == 256 bins
        __syncthreads();
        const uint32_t prefix = sc[0];
        const int      shift  = 24 - pass * 8;
        const uint32_t pmask  = (pass == 0) ? 0u : (0xFFFFFFFFu << (shift + 8));
#pragma unroll 4
        for (int i = tid; i < ROW_D / 4; i += NT) {
            v4f v = row4[i];                // ds_load_b128
#pragma unroll
            for (int c = 0; c < 4; ++c) {
                uint32_t key = order_key(v[c]);
                if ((key & pmask) == prefix)
                    atomicAdd(&hist[(key >> shift) & 0xFFu], 1u);
            }
        }
        __syncthreads();
        if (tid == 0) {
            uint32_t rem = sc[1];
            uint32_t cum = 0u;
            for (int b = 255; b >= 0; --b) {
                uint32_t h = hist[b];
                if (cum + h >= rem) {
                    sc[1] = rem - cum;                       // still needed inside this bucket
                    sc[0] = prefix | ((uint32_t)b << shift); // extend prefix
                    break;
                }
                cum += h;
            }
        }
        __syncthreads();
    }

    const uint32_t T       = sc[0];   // exact key of the k-th largest element
    const uint32_t need_eq = sc[1];   // how many key==T elements to keep (>=1)

    // ---- Emit masked row (b128 NT stores); collect key==T indices ----
#pragma unroll 4
    for (int i = tid; i < ROW_D / 4; i += NT) {
        v4f v = row4[i];                    // ds_load_b128
        v4f o;
#pragma unroll
        for (int c = 0; c < 4; ++c) {
            float    f   = v[c];
            uint32_t key = order_key(f);
            float    r   = 0.0f;
            if (key > T) {
                r = f;
            } else if (key == T) {
                uint32_t slot = atomicAdd(&sc[2], 1u);
                if (slot < EQCAP) eqlist[slot] = (uint32_t)(4 * i + c);
            }
            o[c] = r;
        }
        __builtin_nontemporal_store(o, &outp4[i]);   // global_store_b128 th:NT
    }
    // Ensure all waves' output stores have completed before the fix-up pass
    // may overwrite the same addresses from a different wave.
    __threadfence();
    __syncthreads();

    // ---- Tie fix-up: keep the need_eq lowest-index equal elements ----
    if (tid == 0) {
        uint32_t n = sc[2]; if (n > EQCAP) n = EQCAP;
        if (n <= need_eq) {
            for (uint32_t j = 0; j < n; ++j) {
                uint32_t idx = eqlist[j];
                outp[idx] = row[idx];
            }
        } else {
            for (uint32_t j = 0; j < need_eq; ++j) {
                uint32_t best = 0xFFFFFFFFu, bi = 0u;
                for (uint32_t m = 0; m < n; ++m)
                    if (eqlist[m] < best) { best = eqlist[m]; bi = m; }
                eqlist[bi] = 0xFFFFFFFFu;
                outp[best] = row[best];
            }
        }
    }
}

extern "C" void kernel_launch(void* const* d_in, const int* in_sizes, int n_in,
                              void* d_out, int out_size, void* d_ws, size_t ws_size,
                              hipStream_t stream) {
    (void)n_in; (void)d_ws; (void)ws_size; (void)out_size;
    const float* x   = (const float*)d_in[0];
    float*       out = (float*)d_out;
    const int rows = in_sizes[0] / ROW_D;   // 4096
    dim3 grid((unsigned)rows), block(NT);
    hipLaunchKernelGGL(TopKSparsity_48232482734126_kernel, grid, block,
                       LDS_BYTES, stream, x, out);
}